// TorchLSTMModel_292057776187
// MI455X (gfx1250) — compile-verified
//
#include <hip/hip_runtime.h>
#include <math.h>

typedef __attribute__((ext_vector_type(2))) float v2f;
typedef __attribute__((ext_vector_type(8))) float v8f;

#define T_TOT 4096
#define BB    256
#define II    64
#define HH    36
#define GG    144   // 4*H
#define OO    2
#define NT    9     // GG / 16

__device__ __forceinline__ float sigm(float x) { return 1.0f / (1.0f + expf(-x)); }

// ---------------------------------------------------------------------------
// Phase 1: xg[t0..t0+tc) = x @ W_ih^T + (b_ih + b_hh), fp32 WMMA 16x16x4.
// Output stored WMMA-swizzled: xg[((mtile*9 + ntile)*32 + lane)*8 + v]
// where mtile = chunk-local (t*16 + b/16), so phase 2 loads its C tile as one
// contiguous v8f per lane.
// ---------------------------------------------------------------------------
__global__ __launch_bounds__(128) void lstm_xgemm(
    const float* __restrict__ x,    // [T,B,I]
    const float* __restrict__ Wih,  // [4H,I]
    const float* __restrict__ bih,  // [4H]
    const float* __restrict__ bhh,  // [4H]
    float* __restrict__ xg,
    int t0, int tc)
{
    const int lane  = threadIdx.x & 31;
    const int wv    = threadIdx.x >> 5;       // 0..3
    const int ln    = lane & 15;
    const int half  = lane >> 4;
    const int mtile = blockIdx.x * 4 + wv;    // 0 .. tc*16-1
    const int ntile = blockIdx.y;             // 0..8
    const int n     = ntile * 16 + ln;

    const size_t row = (size_t)mtile * 16 + ln;                 // chunk-local row
    const v2f* xrow = (const v2f*)(x + ((size_t)t0 * BB + row) * II);
    const v2f* wrow = (const v2f*)(Wih + (size_t)n * II);

    v8f acc = {};
#pragma unroll
    for (int kk = 0; kk < II / 4; ++kk) {     // 16 k-steps of 4
        v2f a = xrow[kk * 2 + half];          // x[row, 4kk + 2*half + {0,1}]
        v2f b = wrow[kk * 2 + half];          // Wih[n, 4kk + 2*half + {0,1}]
        acc = __builtin_amdgcn_wmma_f32_16x16x4_f32(
            false, a, false, b, (short)0, acc, false, false);
    }

    const float bias = bih[n] + bhh[n];       // N = n for all 8 accumulator regs
#pragma unroll
    for (int v = 0; v < 8; ++v) acc[v] += bias;

    *(v8f*)(xg + ((size_t)(mtile * NT + ntile) * 32 + lane) * 8) = acc;
}

// ---------------------------------------------------------------------------
// Phase 2: sequential scan. One block per 16 batch rows (16 blocks total, one
// WGP each, no inter-block sync needed). 288 threads = 9 waves; wave w owns
// gate columns [16w, 16w+16). h, c, W_hh live in LDS; 9 chained
// V_WMMA_F32_16X16X4_F32 per wave per step; 2 barriers per step. Output
// projection (O=2) for step t-1 runs concurrently with step t's GEMM.
// ---------------------------------------------------------------------------
__global__ __launch_bounds__(288) void lstm_scan(
    const float* __restrict__ xg,
    const float* __restrict__ Whh,   // [4H,H]
    const float* __restrict__ Wout,  // [O,H]
    const float* __restrict__ bout,  // [O]
    float* __restrict__ hst,         // [B,H] persisted state
    float* __restrict__ cst,         // [B,H]
    float* __restrict__ y,           // [T,B,O]
    int t0, int tc, int first)
{
    __shared__ float sWhh[GG * HH];      // 20736 B
    __shared__ float sH[16 * HH];        //  2304 B
    __shared__ float sC[16 * HH];        //  2304 B
    __shared__ float sG[16 * GG];        //  9216 B
    __shared__ float sWout[OO * HH];
    __shared__ float sBout[OO];

    const int tid  = threadIdx.x;
    const int lane = tid & 31;
    const int wv   = tid >> 5;           // ntile 0..8
    const int ln   = lane & 15;
    const int half = lane >> 4;
    const int b0   = blockIdx.x * 16;
    const int n    = wv * 16 + ln;

    for (int i = tid; i < GG * HH; i += 288) sWhh[i] = Whh[i];
    for (int i = tid; i < OO * HH; i += 288) sWout[i] = Wout[i];
    if (tid < OO) sBout[tid] = bout[tid];
    for (int i = tid; i < 16 * HH; i += 288) {
        sH[i] = first ? 0.0f : hst[(size_t)b0 * HH + i];
        sC[i] = first ? 0.0f : cst[(size_t)b0 * HH + i];
    }
    __syncthreads();

    const v2f* h2 = (const v2f*)sH;      // h[m, k] pairs: idx = m*18 + k/2
    const v2f* w2 = (const v2f*)sWhh;    // Whh[n, k] pairs: idx = n*18 + k/2

    for (int t = 0; t < tc; ++t) {
        // --- gates GEMM: reads sH (step t-1), accumulator tile from xg ---
        const int mtile = t * 16 + (b0 >> 4);
        v8f xv = *(const v8f*)(xg + ((size_t)(mtile * NT + wv) * 32 + lane) * 8);
        v8f acc = {};
#pragma unroll
        for (int kk = 0; kk < NT; ++kk) {     // K = 36 = 9 x 4
            v2f a = h2[ln * (HH / 2) + kk * 2 + half];
            v2f b = w2[n  * (HH / 2) + kk * 2 + half];
            acc = __builtin_amdgcn_wmma_f32_16x16x4_f32(
                false, a, false, b, (short)0, acc, false, false);
        }
#pragma unroll
        for (int v = 0; v < 8; ++v)
            sG[(v + 8 * half) * GG + n] = acc[v] + xv[v];

        // --- output projection for previous step (reads sH, overlaps GEMM) ---
        if (t > 0 && tid < 32) {
            const int m = tid >> 1, o = tid & 1;
            float a = sBout[o];
            for (int j = 0; j < HH; ++j) a += sH[m * HH + j] * sWout[o * HH + j];
            y[((size_t)(t0 + t - 1) * BB + b0 + m) * OO + o] = a;
        }
        __syncthreads();

        // --- elementwise cell update: 576 cells over 288 threads ---
        for (int idx = tid; idx < 16 * HH; idx += 288) {
            const int m = idx / HH, j = idx - m * HH;
            const float ig = sG[m * GG + j];
            const float fg = sG[m * GG + HH + j];
            const float gg = sG[m * GG + 2 * HH + j];
            const float og = sG[m * GG + 3 * HH + j];
            const float cn = sigm(fg) * sC[idx] + sigm(ig) * tanhf(gg);
            sC[idx] = cn;
            sH[idx] = sigm(og) * tanhf(cn);
        }
        __syncthreads();
    }

    // output projection for the chunk's last step
    if (tid < 32) {
        const int m = tid >> 1, o = tid & 1;
        float a = sBout[o];
        for (int j = 0; j < HH; ++j) a += sH[m * HH + j] * sWout[o * HH + j];
        y[((size_t)(t0 + tc - 1) * BB + b0 + m) * OO + o] = a;
    }

    // persist state for next chunk
    for (int i = tid; i < 16 * HH; i += 288) {
        hst[(size_t)b0 * HH + i] = sH[i];
        cst[(size_t)b0 * HH + i] = sC[i];
    }
}

// ---------------------------------------------------------------------------
extern "C" void kernel_launch(void* const* d_in, const int* in_sizes, int n_in,
                              void* d_out, int out_size, void* d_ws, size_t ws_size,
                              hipStream_t stream)
{
    const float* x    = (const float*)d_in[0];
    const float* Wih  = (const float*)d_in[1];
    const float* Whh  = (const float*)d_in[2];
    const float* bih  = (const float*)d_in[3];
    const float* bhh  = (const float*)d_in[4];
    const float* Wout = (const float*)d_in[5];
    const float* bout = (const float*)d_in[6];
    float* y = (float*)d_out;

    float* hst = (float*)d_ws;
    float* cst = hst + (size_t)BB * HH;
    float* xg  = cst + (size_t)BB * HH;     // offset 73728 B, 32B-aligned

    const size_t stateBytes = (size_t)2 * BB * HH * sizeof(float);
    const size_t stepBytes  = (size_t)BB * GG * sizeof(float);   // 147456 B / timestep
    const size_t avail = (ws_size > stateBytes) ? (ws_size - stateBytes) : 0;
    long TC = (long)(avail / stepBytes);
    if (TC < 1) TC = 1;
    if (TC > T_TOT) TC = T_TOT;

    for (int t0 = 0; t0 < T_TOT; t0 += (int)TC) {
        const int tc = (T_TOT - t0 < (int)TC) ? (T_TOT - t0) : (int)TC;
        dim3 g1((unsigned)tc * 4, NT);
        lstm_xgemm<<<g1, 128, 0, stream>>>(x, Wih, bih, bhh, xg, t0, tc);
        lstm_scan<<<16, 288, 0, stream>>>(xg, Whh, Wout, bout, hst, cst, y,
                                          t0, tc, t0 == 0 ? 1 : 0);
    }
}